// Eigenvalues3x3_79654463471929
// MI455X (gfx1250) — compile-verified
//
#include <hip/hip_runtime.h>
#include <stdint.h>

// ---------------------------------------------------------------------------
// Eigenvalues of symmetric 3x3 matrices, one per voxel of a 192^3 volume.
// Memory-bound (48 B/voxel @ 23.3 TB/s => ~15us floor). Strategy:
//   * 256-thread block stages a 1024-voxel tile (36 KB) into LDS using
//     gfx1250 async global->LDS B128 copies (ASYNCcnt / s_wait_asynccnt).
//   * conflict-free LDS transpose (stride-9 and stride-3 are coprime to 64
//     banks) converts coalesced B128 traffic into per-voxel 9-float rows.
//   * analytic eigen formula; one acosf + one hw sin/cos pair instead of
//     three cosf calls (2cos(t±2π/3) = -cos t ∓ √3 sin t).
//   * results drained back via async LDS->global B128 stores; S_ENDPGM's
//     implicit wait-idle covers the final ASYNCcnt.
// WMMA is inapplicable: elementwise op, no GEMM structure, bandwidth bound.
// ---------------------------------------------------------------------------

#define THREADS 256
#define VPB     1024   // voxels per block (1024*36 B = 36 KB staged input)

// Builtin signature (from the round-1 diagnostic): the b128 async copies take
// pointers to int __attribute__((vector_size(16))) in AS1 (global) / AS3 (LDS).
typedef int v4i __attribute__((vector_size(16)));
typedef v4i __attribute__((address_space(1))) gv4i;   // global <4 x i32>
typedef v4i __attribute__((address_space(3))) lv4i;   // LDS    <4 x i32>

#if __has_builtin(__builtin_amdgcn_global_load_async_to_lds_b128) && \
    __has_builtin(__builtin_amdgcn_global_store_async_from_lds_b128)
#define USE_ASYNC_LDS 1
#else
#define USE_ASYNC_LDS 0
#endif

__device__ __forceinline__ void eig3_sym(const float h[9],
                                         float& o0, float& o1, float& o2) {
    // q = trace/3
    const float q = (h[0] + h[4] + h[8]) * (1.0f / 3.0f);
    // deviatoric C = H - q*I
    const float c00 = h[0] - q, c11 = h[4] - q, c22 = h[8] - q;
    const float c01 = h[1], c02 = h[2], c12 = h[5];
    // ||C||_F^2 using all 9 stored entries (input is symmetrized, h[1]==h[3] etc.)
    const float fro2 = c00 * c00 + c11 * c11 + c22 * c22
                     + h[1] * h[1] + h[3] * h[3]
                     + h[2] * h[2] + h[6] * h[6]
                     + h[5] * h[5] + h[7] * h[7];
    const float p  = sqrtf(fro2 * (1.0f / 6.0f));
    const float pp = p + 1e-10f;
    // det(C) for symmetric C; det(B) = det(C/pp) = det(C)/pp^3
    const float detC = c00 * (c11 * c22 - c12 * c12)
                     - c01 * (c01 * c22 - c12 * c02)
                     + c02 * (c01 * c12 - c11 * c02);
    const float inv  = 1.0f / pp;
    const float detB = detC * inv * inv * inv;
    float x = detB * 0.5f;
    x = fminf(1.0f, fmaxf(-1.0f, x));
    const float theta = acosf(x) * (1.0f / 3.0f);
    // three cosines from a single hw sin/cos pair (theta in [0, pi/3])
    const float sn = __sinf(theta);
    const float cs = __cosf(theta);
    const float SQRT3 = 1.7320508075688772935f;
    const float e0 = (2.0f * cs) * p + q;          // k = 0
    const float e1 = (-cs - SQRT3 * sn) * p + q;   // k = 1  (smallest)
    const float e2 = (-cs + SQRT3 * sn) * p + q;   // k = 2
    // exact ascending sort (3-element min/max network)
    const float t0 = fminf(e1, e2), t1 = fmaxf(e1, e2);
    o0 = fminf(t0, e0);
    o1 = fmaxf(t0, fminf(t1, e0));
    o2 = fmaxf(t1, e0);
}

__global__ __launch_bounds__(THREADS)
void Eigenvalues3x3_79654463471929_kernel(const float* __restrict__ H,
                                          float* __restrict__ out,
                                          int nvox) {
    __shared__ __align__(16) float sIn[9 * VPB];   // 36 KB staged Hessians
    __shared__ __align__(16) float sOut[3 * VPB];  // 12 KB staged eigenvalues

    const int  tid   = threadIdx.x;
    const long vbase = (long)blockIdx.x * VPB;

    if (vbase + VPB <= (long)nvox) {
        const float* gsrc = H + vbase * 9;
        float*       gdst = out + vbase * 3;

        // ---- stage input tile: 9 coalesced per-lane B128 copies -----------
#if USE_ASYNC_LDS
        {
            const uint32_t  lin = (uint32_t)(uintptr_t)(void*)&sIn[0];
            const uintptr_t gin = (uintptr_t)gsrc;
#pragma unroll
            for (int i = 0; i < 9; ++i) {
                const uint32_t e = (uint32_t)(i * THREADS + tid) * 16u;
                __builtin_amdgcn_global_load_async_to_lds_b128(
                    (gv4i*)(gin + (uintptr_t)e),
                    (lv4i*)(lin + e),
                    0, 0);
            }
        }
#if __has_builtin(__builtin_amdgcn_s_wait_asynccnt)
        __builtin_amdgcn_s_wait_asynccnt(0);
#else
        asm volatile("s_wait_asynccnt 0" ::: "memory");
#endif
        __syncthreads();
#else
#pragma unroll
        for (int i = 0; i < 9; ++i) {
            ((float4*)sIn)[i * THREADS + tid] =
                ((const float4*)gsrc)[i * THREADS + tid];
        }
        __syncthreads();
#endif

        // ---- compute 4 voxels per thread (stride-THREADS => bank-clean) ---
#pragma unroll
        for (int j = 0; j < 4; ++j) {
            const int v = j * THREADS + tid;
            float h[9];
#pragma unroll
            for (int k = 0; k < 9; ++k) h[k] = sIn[v * 9 + k];  // stride 9: no conflicts
            float a, b, c;
            eig3_sym(h, a, b, c);
            sOut[v * 3 + 0] = a;   // stride 3: no conflicts
            sOut[v * 3 + 1] = b;
            sOut[v * 3 + 2] = c;
        }
        __syncthreads();

        // ---- drain output tile: 3 coalesced per-lane B128 copies ----------
#if USE_ASYNC_LDS
        {
            const uint32_t  lou = (uint32_t)(uintptr_t)(void*)&sOut[0];
            const uintptr_t gou = (uintptr_t)gdst;
#pragma unroll
            for (int i = 0; i < 3; ++i) {
                const uint32_t e = (uint32_t)(i * THREADS + tid) * 16u;
                __builtin_amdgcn_global_store_async_from_lds_b128(
                    (gv4i*)(gou + (uintptr_t)e),
                    (lv4i*)(lou + e),
                    0, 0);
            }
            // S_ENDPGM performs an implicit wait-idle (ASYNCcnt -> 0).
        }
#else
#pragma unroll
        for (int i = 0; i < 3; ++i) {
            ((float4*)gdst)[i * THREADS + tid] =
                ((const float4*)sOut)[i * THREADS + tid];
        }
#endif
    } else {
        // generic tail path (not taken for 192^3, which is divisible by VPB)
        for (long v = vbase + tid; v < (long)nvox; v += THREADS) {
            float h[9];
            for (int k = 0; k < 9; ++k) h[k] = H[v * 9 + k];
            float a, b, c;
            eig3_sym(h, a, b, c);
            out[v * 3 + 0] = a;
            out[v * 3 + 1] = b;
            out[v * 3 + 2] = c;
        }
    }
}

extern "C" void kernel_launch(void* const* d_in, const int* in_sizes, int n_in,
                              void* d_out, int out_size, void* d_ws, size_t ws_size,
                              hipStream_t stream) {
    const float* H   = (const float*)d_in[0];
    float*       out = (float*)d_out;
    const long nvox  = (long)in_sizes[0] / 9;      // 192^3 = 7,077,888
    const int  nblk  = (int)((nvox + VPB - 1) / VPB);  // 6912 blocks
    Eigenvalues3x3_79654463471929_kernel<<<nblk, THREADS, 0, stream>>>(
        H, out, (int)nvox);
}